// Int4Embedding_40905268527134
// MI455X (gfx1250) — compile-verified
//
#include <hip/hip_runtime.h>
#include <hip/hip_bf16.h>
#include <stdint.h>

// Int4 embedding gather + dequant for MI455X (gfx1250, wave32).
//
//   ids          : [4,4096] int32            (16384 tokens)
//   weight_packed: [128000, 1024] uint8      (2 nibbles/byte, high nibble first)
//   scale        : [128000, 16]   f32        (group size 128)
//   zero_point   : [128000, 16]   f32
//   out          : [16384, 2048]  f32
//
// Perf model: 134 MB of output stores dominate (~5.8 us @ 23.3 TB/s). The whole
// quantized table (147 MB) fits in the 192 MB L2, so gather reads are L2 hits
// as long as the output stream does not evict it -> output uses TH=NT stores.

#define EMB_DIM   2048
#define GROUPS    16          // EMB_DIM / 128
#define ROW_BYTES 1024        // EMB_DIM / 2
#define LANES_PER_TOKEN 64    // each lane: 16 packed bytes -> 32 floats

typedef float    v4f  __attribute__((ext_vector_type(4)));
typedef uint32_t v4u  __attribute__((ext_vector_type(4)));

__global__ void __launch_bounds__(256)
int4emb_dequant_kernel(const int* __restrict__ ids,
                       const uint8_t* __restrict__ wpacked,
                       const float* __restrict__ scale,
                       const float* __restrict__ zp,
                       float* __restrict__ out,
                       int n_tokens)
{
    const int g    = blockIdx.x * 256 + threadIdx.x;
    const int tok  = g >> 6;            // token position in [0, n_tokens)
    const int lane = g & (LANES_PER_TOKEN - 1);
    if (tok >= n_tokens) return;

    // Row id is uniform across the wave (64 threads/token, wave32) ->
    // move it to an SGPR so weight/scale loads take the saddr form.
    int row = ids[tok];
    row = __builtin_amdgcn_readfirstlane(row);

    // --- load 16 packed bytes (one global_load_b128), L2-resident, RT policy
    const v4u* wp = (const v4u*)(wpacked + (size_t)row * ROW_BYTES) + lane;
    const v4u w = *wp;

    // --- per-thread quant group: elements [lane*32, lane*32+32) all fall in
    //     group lane/4 (32 divides 128)
    const int grp = lane >> 2;
    const float s = scale[row * GROUPS + grp];
    const float c = -zp[row * GROUPS + grp] * s;   // out = q*s + c

    float o[32];
#pragma unroll
    for (int i = 0; i < 4; ++i) {
        const uint32_t v  = w[i];
        // split nibbles; element order per byte b: (b>>4) then (b&15)
        const uint32_t hi = (v >> 4) & 0x0F0F0F0Fu;
        const uint32_t lo =  v       & 0x0F0F0F0Fu;
        float* e = &o[8 * i];
#pragma unroll
        for (int b = 0; b < 4; ++b) {
            // backend folds these to v_cvt_f32_ubyte{0..3}
            const float qh = (float)((hi >> (8 * b)) & 0xFFu);
            const float ql = (float)((lo >> (8 * b)) & 0xFFu);
            e[2 * b + 0] = fmaf(qh, s, c);
            e[2 * b + 1] = fmaf(ql, s, c);
        }
    }

    // --- 128 contiguous bytes per lane, non-temporal so the 134 MB output
    //     stream does not evict the L2-resident weight table.
    v4f* op = (v4f*)(out + (size_t)tok * EMB_DIM + lane * 32);
#pragma unroll
    for (int i = 0; i < 8; ++i) {
        v4f q = { o[4 * i + 0], o[4 * i + 1], o[4 * i + 2], o[4 * i + 3] };
        __builtin_nontemporal_store(q, op + i);
    }
}

extern "C" void kernel_launch(void* const* d_in, const int* in_sizes, int n_in,
                              void* d_out, int out_size, void* d_ws, size_t ws_size,
                              hipStream_t stream)
{
    const int*      ids     = (const int*)d_in[0];
    const uint8_t*  wpacked = (const uint8_t*)d_in[1];
    const float*    scale   = (const float*)d_in[2];
    const float*    zp      = (const float*)d_in[3];
    float*          out     = (float*)d_out;

    const int n_tokens = in_sizes[0];                       // B*S = 16384
    const long long total_threads = (long long)n_tokens * LANES_PER_TOKEN;
    const int blocks = (int)((total_threads + 255) / 256);

    int4emb_dequant_kernel<<<blocks, 256, 0, stream>>>(
        ids, wpacked, scale, zp, out, n_tokens);
}